// EpiNetModel_82858509074939
// MI455X (gfx1250) — compile-verified
//
#include <hip/hip_runtime.h>
#include <hip/hip_bf16.h>
#include <math.h>

// ---- problem constants (match reference) ----
constexpr int kB    = 256;
constexpr int kD    = 256;
constexpr int kN    = 200000;
constexpr int kK    = 32;
constexpr int kIN   = 784;
constexpr int kHID  = 512;
constexpr int kNCLS = 100;
constexpr float kDECAY  = 0.01f;
constexpr float kLAMBDA = 0.5f;
constexpr float kEPS    = 1e-8f;

// chunking for the scores/top-k stage
constexpr int kNC = 512;                       // columns per chunk
constexpr int kCC = (kN + kNC - 1) / kNC;      // 391 chunks

typedef float    v2f __attribute__((ext_vector_type(2)));
typedef float    v8f __attribute__((ext_vector_type(8)));
typedef unsigned v4u __attribute__((ext_vector_type(4)));
typedef unsigned v8u __attribute__((ext_vector_type(8)));

__device__ __forceinline__ v8f wmma_f32_4(v2f a, v2f b, v8f c) {
  return __builtin_amdgcn_wmma_f32_16x16x4_f32(false, a, false, b, (short)0, c,
                                               false, false);
}

// ---------------------------------------------------------------------------
// 1) encoder: z = x @ enc_w + enc_b   [256,784]x[784,256]
//    grid (16,16) tiles, 1 wave / tile
// ---------------------------------------------------------------------------
__global__ __launch_bounds__(32) void k_encoder(const float* __restrict__ x,
                                                const float* __restrict__ w,
                                                const float* __restrict__ bias,
                                                float* __restrict__ z) {
  const int mt = blockIdx.x, nt = blockIdx.y;
  const int lane = threadIdx.x;
  const int m = lane & 15, kh = lane >> 4;
  const float* arow = x + (size_t)(mt * 16 + m) * kIN;
  v8f acc = {};
  for (int k0 = 0; k0 < kIN; k0 += 4) {
    const int k = k0 + kh * 2;
    v2f a, b;
    a.x = arow[k];
    a.y = arow[k + 1];
    b.x = w[(size_t)k * kD + nt * 16 + m];
    b.y = w[(size_t)(k + 1) * kD + nt * 16 + m];
    acc = wmma_f32_4(a, b, acc);
  }
  const float bn = bias[nt * 16 + m];
#pragma unroll
  for (int v = 0; v < 8; ++v) {
    const int mm = mt * 16 + v + 8 * kh;
    z[(size_t)mm * kD + nt * 16 + m] = acc[v] + bn;
  }
}

// ---------------------------------------------------------------------------
// 2a) inverse z-row norms: zinv[b] = 1/max(||z_b||, eps)  (wave per row)
// ---------------------------------------------------------------------------
__global__ __launch_bounds__(256) void k_zinv(const float* __restrict__ z,
                                              float* __restrict__ zinv) {
  const int wave = threadIdx.x >> 5, lane = threadIdx.x & 31;
  const int row = blockIdx.x * 8 + wave;
  if (row >= kB) return;
  const float* zr = z + (size_t)row * kD;
  float s = 0.f;
#pragma unroll
  for (int i = 0; i < kD / 32; ++i) {
    const float v = zr[lane + i * 32];
    s += v * v;
  }
#pragma unroll
  for (int off = 16; off > 0; off >>= 1) s += __shfl_xor(s, off, 32);
  if (lane == 0) zinv[row] = 1.0f / fmaxf(sqrtf(s), kEPS);
}

// ---------------------------------------------------------------------------
// 2b) per-memory factor: sal[n] = r0*exp(-decay*tau); fac[n] = sal/max(||c_n||,eps)
// ---------------------------------------------------------------------------
__global__ __launch_bounds__(256) void k_factor(const float* __restrict__ cbuf,
                                                const float* __restrict__ r0,
                                                const float* __restrict__ tau,
                                                float* __restrict__ fac,
                                                float* __restrict__ sal) {
  const int wave = threadIdx.x >> 5, lane = threadIdx.x & 31;
  const int n = blockIdx.x * 8 + wave;
  if (n >= kN) return;
  const float* cr = cbuf + (size_t)n * kD;
  float s = 0.f;
#pragma unroll
  for (int i = 0; i < kD / 32; ++i) {
    const float v = cr[lane + i * 32];
    s += v * v;
  }
#pragma unroll
  for (int off = 16; off > 0; off >>= 1) s += __shfl_xor(s, off, 32);
  if (lane == 0) {
    const float sv = r0[n] * __expf(-kDECAY * tau[n]);
    sal[n] = sv;
    fac[n] = sv / fmaxf(sqrtf(s), kEPS);
  }
}

// ---------------------------------------------------------------------------
// 3) scores tile (WMMA f32) + per-row local top-32 within a 512-column chunk
//    z row tile staged into LDS by the Tensor Data Mover (wave 0 issues it).
//    grid (kCC, 16 row-tiles), 256 threads (8 waves)
// ---------------------------------------------------------------------------
__global__ __launch_bounds__(256) void k_scores_topk(
    const float* __restrict__ z, const float* __restrict__ zinv,
    const float* __restrict__ cbuf, const float* __restrict__ fac,
    float* __restrict__ cand_val, int* __restrict__ cand_idx) {
  __shared__ float sZ[16 * kD];   // 16 KB: z row tile
  __shared__ float sS[16 * kNC];  // 32 KB: scaled scores tile
  const int chunk = blockIdx.x;
  const int rt = blockIdx.y;
  const int tid = threadIdx.x;

  // ---- TDM: DMA the contiguous 16x256 f32 z tile (16 KB) into LDS ----
  if (tid < 32) {
    const unsigned lds = (unsigned)(size_t)&sZ[0];  // generic->LDS offset (low 32b)
    const unsigned long long ga =
        (unsigned long long)(const void*)(z + (size_t)(rt * 16) * kD);
    constexpr unsigned kElems = 16u * kD;  // 4096 f32 elements, 1-D tile
    v4u g0;
    g0[0] = 1u;                                   // count=1, user descriptor
    g0[1] = lds;                                  // lds_addr (bytes)
    g0[2] = (unsigned)ga;                         // global_addr[31:0]
    g0[3] = (unsigned)(ga >> 32) | (2u << 30);    // global_addr[56:32] | type=2
    v8u g1;
    g1[0] = 2u << 16;                             // data_size = 4 bytes
    g1[1] = (kElems & 0xFFFFu) << 16;             // tensor_dim0[15:0]
    g1[2] = (kElems >> 16);                       // tensor_dim0[31:16]
    g1[3] = (kElems & 0xFFFFu) << 16;             // tile_dim0 = 4096
    g1[4] = 0u;                                   // tile_dim1/2 unused
    g1[5] = kElems;                               // tensor_dim0_stride[31:0]
    g1[6] = 0u;
    g1[7] = 0u;
    asm volatile("tensor_load_to_lds %0, %1\n\t"
                 "s_wait_tensorcnt 0x0"
                 :: "s"(g0), "s"(g1) : "memory");
  }
  __syncthreads();

  const int lane = tid & 31, wave = tid >> 5;
  const int m = lane & 15, kh = lane >> 4;
  const int nbase = chunk * kNC;

  for (int j = 0; j < (kNC / 16) / 8; ++j) {
    const int ct = wave * 4 + j;
    const int nl = ct * 16 + m;   // local column
    const int ng = nbase + nl;    // global column
    const bool valid = ng < kN;
    const float* cp = cbuf + (size_t)(valid ? ng : 0) * kD;
    if (valid) __builtin_prefetch(cp, 0, 3);  // global_prefetch_b8
    v8f acc = {};
    for (int k0 = 0; k0 < kD; k0 += 4) {
      const int k = k0 + kh * 2;
      v2f a, b;
      a.x = sZ[m * kD + k];
      a.y = sZ[m * kD + k + 1];
      const float2 bb = *(const float2*)(cp + k);
      b.x = valid ? bb.x : 0.f;
      b.y = valid ? bb.y : 0.f;
      acc = wmma_f32_4(a, b, acc);
    }
    const float fcol = valid ? fac[ng] : 0.f;
#pragma unroll
    for (int v = 0; v < 8; ++v) {
      const int mm = v + 8 * kh;
      const float sv = acc[v] * zinv[rt * 16 + mm] * fcol;
      sS[mm * kNC + nl] = valid ? sv : -__builtin_inff();
    }
  }
  __syncthreads();

  // per-row local top-32: each wave handles two rows
  for (int rr = wave * 2; rr < wave * 2 + 2; ++rr) {
    float* row = sS + rr * kNC;
    const int grow = rt * 16 + rr;
    for (int sel = 0; sel < kK; ++sel) {
      float best = -__builtin_inff();
      int bidx = -1;
#pragma unroll
      for (int q0 = 0; q0 < kNC / 32; ++q0) {
        const int q = lane + q0 * 32;
        const float v = row[q];
        if (v > best) { best = v; bidx = q; }
      }
#pragma unroll
      for (int off = 16; off > 0; off >>= 1) {
        const float ov = __shfl_xor(best, off, 32);
        const int oi = __shfl_xor(bidx, off, 32);
        if (ov > best) { best = ov; bidx = oi; }
      }
      if (lane == 0) {
        const size_t slot = ((size_t)grow * kCC + chunk) * kK + sel;
        cand_val[slot] = best;
        cand_idx[slot] = (bidx >= 0) ? (nbase + bidx) : 0;
        if (bidx >= 0) row[bidx] = -__builtin_inff();
      }
      asm volatile("s_wait_dscnt 0x0" ::: "memory");  // wave-sync the LDS RMW
    }
  }
}

// ---------------------------------------------------------------------------
// 4) merge candidates -> global top-32, softmax weights, recall vector r
//    one block (256 threads) per batch row
// ---------------------------------------------------------------------------
__global__ __launch_bounds__(256) void k_merge(
    float* __restrict__ cand_val, const int* __restrict__ cand_idx,
    const float* __restrict__ sal, float* __restrict__ wgt,
    int* __restrict__ topi, float* __restrict__ salk,
    const float* __restrict__ zbuf, float* __restrict__ rout) {
  __shared__ float red_v[256];
  __shared__ int red_i[256];
  __shared__ float topv_s[kK];
  __shared__ int topi_s[kK];
  __shared__ float w_s[kK];
  const int b = blockIdx.x, tid = threadIdx.x;
  const int M = kCC * kK;
  float* cv = cand_val + (size_t)b * M;
  const int* ci = cand_idx + (size_t)b * M;

  for (int sel = 0; sel < kK; ++sel) {
    float best = -__builtin_inff();
    int bp = -1;
    for (int q = tid; q < M; q += 256) {
      const float v = cv[q];
      if (v > best) { best = v; bp = q; }
    }
    red_v[tid] = best;
    red_i[tid] = bp;
    __syncthreads();
    for (int s = 128; s > 0; s >>= 1) {
      if (tid < s && red_v[tid + s] > red_v[tid]) {
        red_v[tid] = red_v[tid + s];
        red_i[tid] = red_i[tid + s];
      }
      __syncthreads();
    }
    if (tid == 0) {
      const int p = red_i[0];
      topv_s[sel] = red_v[0];
      topi_s[sel] = (p >= 0) ? ci[p] : 0;
      if (p >= 0) cv[p] = -__builtin_inff();
    }
    __syncthreads();
  }

  if (tid < kK) {
    const float v = topv_s[tid];
    float mx = v;
#pragma unroll
    for (int off = 16; off > 0; off >>= 1) mx = fmaxf(mx, __shfl_xor(mx, off, 32));
    const float e = __expf(v - mx);
    float s = e;
#pragma unroll
    for (int off = 16; off > 0; off >>= 1) s += __shfl_xor(s, off, 32);
    const float w = e / s;
    w_s[tid] = w;
    wgt[b * kK + tid] = w;
    const int ix = topi_s[tid];
    topi[b * kK + tid] = ix;
    salk[b * kK + tid] = sal[ix];
  }
  __syncthreads();

  // r[b, d] = sum_k w[k] * z_buffer[idx[k], d]   (d = tid)
  float acc = 0.f;
#pragma unroll 4
  for (int k = 0; k < kK; ++k)
    acc += w_s[k] * zbuf[(size_t)topi_s[k] * kD + tid];
  rout[(size_t)b * kD + tid] = acc;
}

// ---------------------------------------------------------------------------
// 5a) task hidden: h = relu([z|r] @ w1 + b1)   [256,512]x[512,512], WMMA
// ---------------------------------------------------------------------------
__global__ __launch_bounds__(32) void k_dec1(const float* __restrict__ z,
                                             const float* __restrict__ r,
                                             const float* __restrict__ w1,
                                             const float* __restrict__ b1,
                                             float* __restrict__ h) {
  const int mt = blockIdx.x, nt = blockIdx.y;
  const int lane = threadIdx.x;
  const int m = lane & 15, kh = lane >> 4;
  const float* zr = z + (size_t)(mt * 16 + m) * kD;
  const float* rr = r + (size_t)(mt * 16 + m) * kD;
  v8f acc = {};
  for (int k0 = 0; k0 < 2 * kD; k0 += 4) {
    const int k = k0 + kh * 2;
    const float* src = (k0 < kD) ? zr : (rr - kD);  // concat select
    v2f a, b;
    a.x = src[k];
    a.y = src[k + 1];
    b.x = w1[(size_t)k * kHID + nt * 16 + m];
    b.y = w1[(size_t)(k + 1) * kHID + nt * 16 + m];
    acc = wmma_f32_4(a, b, acc);
  }
  const float bn = b1[nt * 16 + m];
#pragma unroll
  for (int v = 0; v < 8; ++v) {
    const int mm = mt * 16 + v + 8 * kh;
    h[(size_t)mm * kHID + nt * 16 + m] = fmaxf(acc[v] + bn, 0.f);
  }
}

// ---------------------------------------------------------------------------
// 5b) task logits + mean cross-entropy (atomic accumulate into out[0])
// ---------------------------------------------------------------------------
__global__ __launch_bounds__(128) void k_task_loss(const float* __restrict__ h,
                                                   const float* __restrict__ w2,
                                                   const float* __restrict__ b2,
                                                   const int* __restrict__ y,
                                                   float* __restrict__ out) {
  __shared__ float sl[kNCLS];
  __shared__ float red[128];
  const int b = blockIdx.x, tid = threadIdx.x;
  const float* hr = h + (size_t)b * kHID;
  float logit = 0.f;
  if (tid < kNCLS) {
    logit = b2[tid];
    for (int j = 0; j < kHID; ++j) logit += hr[j] * w2[(size_t)j * kNCLS + tid];
    sl[tid] = logit;
  }
  __syncthreads();
  red[tid] = (tid < kNCLS) ? logit : -__builtin_inff();
  __syncthreads();
  for (int s = 64; s > 0; s >>= 1) {
    if (tid < s) red[tid] = fmaxf(red[tid], red[tid + s]);
    __syncthreads();
  }
  const float mx = red[0];
  __syncthreads();
  red[tid] = (tid < kNCLS) ? __expf(logit - mx) : 0.f;
  __syncthreads();
  for (int s = 64; s > 0; s >>= 1) {
    if (tid < s) red[tid] += red[tid + s];
    __syncthreads();
  }
  if (tid == 0) {
    const float lse = mx + logf(red[0]);
    const float ce = lse - sl[y[b]];
    atomicAdd(out, ce * (1.0f / kB));
  }
}

// ---------------------------------------------------------------------------
// 6a) replay hidden: hm = relu(z_buffer[topidx] @ w1[0:256,:] + b1)  WMMA
//     (zero recall half of the concat contributes nothing)
//     A tile gathered into LDS with per-lane async global->LDS b128 copies.
// ---------------------------------------------------------------------------
__global__ __launch_bounds__(256) void k_replay_h(const int* __restrict__ topi,
                                                  const float* __restrict__ zbuf,
                                                  const float* __restrict__ w1,
                                                  const float* __restrict__ b1,
                                                  float* __restrict__ hm) {
  __shared__ float sA[16 * kD];  // 16 KB gathered A tile
  __shared__ int sIdx[16];
  const int rt = blockIdx.x;
  const int tid = threadIdx.x;
  if (tid < 16) sIdx[tid] = topi[rt * 16 + tid];
  __syncthreads();
  // async gather: each thread issues 4x b128 global->LDS transfers
  for (int e = tid * 4; e < 16 * kD; e += 256 * 4) {
    const int i = e >> 8;        // row within tile (kD == 256)
    const int k = e & (kD - 1);  // column (multiple of 4 -> 16B aligned)
    const unsigned long long g =
        (unsigned long long)(const void*)(zbuf + (size_t)sIdx[i] * kD + k);
    const unsigned loff = (unsigned)(size_t)&sA[e];
    asm volatile("global_load_async_to_lds_b128 %0, %1, off"
                 :: "v"(loff), "v"(g) : "memory");
  }
  asm volatile("s_wait_asynccnt 0x0" ::: "memory");
  __syncthreads();

  const int lane = tid & 31, wave = tid >> 5;
  const int m = lane & 15, kh = lane >> 4;
  for (int j = 0; j < 4; ++j) {
    const int nt = wave * 4 + j;
    const int n = nt * 16 + m;
    v8f acc = {};
    for (int k0 = 0; k0 < kD; k0 += 4) {
      const int k = k0 + kh * 2;
      v2f a, b;
      a.x = sA[m * kD + k];
      a.y = sA[m * kD + k + 1];
      b.x = w1[(size_t)k * kHID + n];
      b.y = w1[(size_t)(k + 1) * kHID + n];
      acc = wmma_f32_4(a, b, acc);
    }
    const float bn = b1[n];
#pragma unroll
    for (int v = 0; v < 8; ++v) {
      const int mm = rt * 16 + v + 8 * kh;
      hm[(size_t)mm * kHID + n] = fmaxf(acc[v] + bn, 0.f);
    }
  }
}

// ---------------------------------------------------------------------------
// 6b) replay logits + weighted CE: out += LAMBDA * sal_topk * ce
// ---------------------------------------------------------------------------
__global__ __launch_bounds__(128) void k_replay_loss(
    const float* __restrict__ hm, const float* __restrict__ w2,
    const float* __restrict__ b2, const int* __restrict__ topi,
    const int* __restrict__ ybuf, const float* __restrict__ salk,
    float* __restrict__ out) {
  __shared__ float sl[kNCLS];
  __shared__ float red[128];
  const int rr = blockIdx.x, tid = threadIdx.x;
  const float* hr = hm + (size_t)rr * kHID;
  float logit = 0.f;
  if (tid < kNCLS) {
    logit = b2[tid];
    for (int j = 0; j < kHID; ++j) logit += hr[j] * w2[(size_t)j * kNCLS + tid];
    sl[tid] = logit;
  }
  __syncthreads();
  red[tid] = (tid < kNCLS) ? logit : -__builtin_inff();
  __syncthreads();
  for (int s = 64; s > 0; s >>= 1) {
    if (tid < s) red[tid] = fmaxf(red[tid], red[tid + s]);
    __syncthreads();
  }
  const float mx = red[0];
  __syncthreads();
  red[tid] = (tid < kNCLS) ? __expf(logit - mx) : 0.f;
  __syncthreads();
  for (int s = 64; s > 0; s >>= 1) {
    if (tid < s) red[tid] += red[tid + s];
    __syncthreads();
  }
  if (tid == 0) {
    const float lse = mx + logf(red[0]);
    const int label = ybuf[topi[rr]];
    const float ce = lse - sl[label];
    atomicAdd(out, kLAMBDA * salk[rr] * ce);
  }
}

// ---------------------------------------------------------------------------
extern "C" void kernel_launch(void* const* d_in, const int* in_sizes, int n_in,
                              void* d_out, int out_size, void* d_ws,
                              size_t ws_size, hipStream_t stream) {
  const float* x     = (const float*)d_in[0];
  const int*   y     = (const int*)d_in[1];
  const float* zbuf  = (const float*)d_in[2];
  const float* cbuf  = (const float*)d_in[3];
  const float* r0    = (const float*)d_in[4];
  const float* tau   = (const float*)d_in[5];
  const int*   ybuf  = (const int*)d_in[6];
  const float* enc_w = (const float*)d_in[7];
  const float* enc_b = (const float*)d_in[8];
  const float* w1    = (const float*)d_in[9];
  const float* b1    = (const float*)d_in[10];
  const float* w2    = (const float*)d_in[11];
  const float* b2    = (const float*)d_in[12];
  float* out = (float*)d_out;

  char* p = (char*)d_ws;
  auto carve = [&](size_t bytes) {
    char* q = p;
    p += (bytes + 255) & ~(size_t)255;
    return q;
  };
  float* ws_z    = (float*)carve((size_t)kB * kD * 4);
  float* ws_zinv = (float*)carve((size_t)kB * 4);
  float* ws_fac  = (float*)carve((size_t)kN * 4);
  float* ws_sal  = (float*)carve((size_t)kN * 4);
  float* ws_cv   = (float*)carve((size_t)kB * kCC * kK * 4);
  int*   ws_ci   = (int*)carve((size_t)kB * kCC * kK * 4);
  int*   ws_topi = (int*)carve((size_t)kB * kK * 4);
  float* ws_w    = (float*)carve((size_t)kB * kK * 4);
  float* ws_salk = (float*)carve((size_t)kB * kK * 4);
  float* ws_r    = (float*)carve((size_t)kB * kD * 4);
  float* ws_h    = (float*)carve((size_t)kB * kHID * 4);
  float* ws_hm   = (float*)carve((size_t)kB * kK * kHID * 4);
  (void)ws_size; (void)in_sizes; (void)n_in; (void)out_size;

  hipMemsetAsync(out, 0, sizeof(float), stream);

  k_encoder<<<dim3(kB / 16, kD / 16), 32, 0, stream>>>(x, enc_w, enc_b, ws_z);
  k_zinv<<<kB / 8, 256, 0, stream>>>(ws_z, ws_zinv);
  k_factor<<<(kN + 7) / 8, 256, 0, stream>>>(cbuf, r0, tau, ws_fac, ws_sal);
  k_scores_topk<<<dim3(kCC, kB / 16), 256, 0, stream>>>(ws_z, ws_zinv, cbuf,
                                                        ws_fac, ws_cv, ws_ci);
  k_merge<<<kB, 256, 0, stream>>>(ws_cv, ws_ci, ws_sal, ws_w, ws_topi, ws_salk,
                                  zbuf, ws_r);
  k_dec1<<<dim3(kB / 16, kHID / 16), 32, 0, stream>>>(ws_z, ws_r, w1, b1, ws_h);
  k_task_loss<<<kB, 128, 0, stream>>>(ws_h, w2, b2, y, out);
  k_replay_h<<<(kB * kK) / 16, 256, 0, stream>>>(ws_topi, zbuf, w1, b1, ws_hm);
  k_replay_loss<<<kB * kK, 128, 0, stream>>>(ws_hm, w2, b2, ws_topi, ybuf,
                                             ws_salk, out);
}